// MultiHeadEVGNetwork_18159121728073
// MI455X (gfx1250) — compile-verified
//
#include <hip/hip_runtime.h>
#include <hip/hip_bf16.h>
#include <math.h>

// ---------------------------------------------------------------------------
// MultiHeadEVG: N=200000, D=256, HIDDEN=256, OUT=256, H=8, hd=32, TOPK=16
//
// Roofline: after folding Q into Wk (logits = E @ Wfold, Wfold = 256x8 padded
// to 256x16) the kernel is one 204.8 MB stream of entity_embeddings
// (~8.8us @ 23.3 TB/s). The skinny GEMM is done with V_WMMA_F32_16X16X4_F32
// (exact fp32) so the hot loop runs on the matrix pipe. V is computed only at
// the 128 selected rows; bk cancels inside softmax and is dropped.
// ---------------------------------------------------------------------------

#define NROWS   200000
#define DCOLS   256
#define NHEADS  8
#define NPAD    16          // WMMA N dimension (heads padded 8 -> 16)
#define TOPK    16
#define NTILES  (NROWS / 16)   // 12500, exact

typedef float v2f  __attribute__((ext_vector_type(2)));
typedef float v4f  __attribute__((ext_vector_type(4)));
typedef float v8f  __attribute__((ext_vector_type(8)));

// ---------------------------------------------------------------------------
// Kernel 0: fold.  Q = Wq@ce + bq;  Wfold[d][h] = sum_j Q[h*32+j]*Wk[h*32+j][d]
// Stored in WMMA-B fragment layout: for k-pair p=d/2, lane-col n, element
// e=d&1:  wfoldB[p*32 + n*2 + e] = Wfold[d][n]  (n>=8 zero-padded)
// ---------------------------------------------------------------------------
__global__ __launch_bounds__(256) void fold_kernel(
    const float* __restrict__ ce, const float* __restrict__ Wq,
    const float* __restrict__ bq, const float* __restrict__ Wk,
    float* __restrict__ wfoldB)
{
    __shared__ float ces[DCOLS];
    __shared__ float Qs[DCOLS];
    const int tid = threadIdx.x;

    ces[tid] = ce[tid];
    __syncthreads();

    // Q[tid] = Wq[tid,:] . ce + bq[tid]
    float q = bq[tid];
    const float* wr = Wq + (size_t)tid * DCOLS;
    #pragma unroll 4
    for (int c = 0; c < DCOLS; ++c) q += wr[c] * ces[c];
    Qs[tid] = q;
    __syncthreads();

    const int d = tid;                 // column of Wfold
    #pragma unroll
    for (int n = 0; n < NPAD; ++n) {
        float s = 0.0f;
        if (n < NHEADS) {
            // Wk[(n*32+j)][d]: for fixed row, d==tid -> coalesced reads
            for (int j = 0; j < 32; ++j)
                s += Qs[n * 32 + j] * Wk[(size_t)(n * 32 + j) * DCOLS + d];
        }
        wfoldB[(d >> 1) * (NPAD * 2) + n * 2 + (d & 1)] = s;
    }
}

// ---------------------------------------------------------------------------
// Kernel 1: the hot streaming pass.  Each wave: one 16-row tile of E,
// 64 x V_WMMA_F32_16X16X4_F32 over K=256, store 16x8 logits (head-major).
// A layout (16x4 f32): lane = m + 16*kp; v0 = E[m][kb+2kp], v1 = E[m][kb+2kp+1]
// B layout (4x16 f32): lane = n + 16*kp; v0 = Wfold[kb+2kp][n], v1 = ...+1
// C/D (16x16 f32): vgpr j: lanes 0-15 -> M=j, lanes 16-31 -> M=8+j
// ---------------------------------------------------------------------------
__global__ __launch_bounds__(256) void logits_wmma_kernel(
    const float* __restrict__ E, const float* __restrict__ wfoldB,
    float* __restrict__ logits)
{
    __shared__ float Bsh[DCOLS * NPAD];   // 16 KB folded-B in fragment order

    const int tid = threadIdx.x;
    for (int i = tid; i < DCOLS * NPAD; i += 256) Bsh[i] = wfoldB[i];
    __syncthreads();

    const int wave = tid >> 5;
    const int lane = tid & 31;
    const int mn   = lane & 15;          // row (A) / col (B) within tile
    const int kp   = lane >> 4;          // k-pair half select

    const int tile = blockIdx.x * 8 + wave;   // wave-uniform
    if (tile >= NTILES) return;               // uniform exit: EXEC stays full

    const int row0 = tile * 16;
    const float* aptr = E + (size_t)(row0 + mn) * DCOLS + 2 * kp;
    const float* bptr = Bsh + (kp * 16 + mn) * 2;   // +64 floats per k-step

    v8f acc = {};
    #pragma unroll 8
    for (int t = 0; t < 64; ++t) {
        v2f a = *(const v2f*)(aptr + t * 4);    // global_load_b64
        v2f b = *(const v2f*)(bptr + t * 64);   // ds_load_b64
        acc = __builtin_amdgcn_wmma_f32_16x16x4_f32(
                  false, a, false, b, (short)0, acc, false, false);
    }

    if (mn < NHEADS) {                          // lane's column == head id
        const int rbase = row0 + (kp ? 8 : 0);  // acc[j] -> row rbase+j
        float* o = logits + (size_t)mn * NROWS + rbase;
        v4f lo = { acc[0], acc[1], acc[2], acc[3] };
        v4f hi = { acc[4], acc[5], acc[6], acc[7] };
        *(v4f*)(o)     = lo;
        *(v4f*)(o + 4) = hi;
    }
}

// ---------------------------------------------------------------------------
// Kernel 2: per-head (one block per head) online softmax stats + exact top-16.
// Per-thread sorted top-16 list -> LDS merge tree.  Tie-break: lower index
// wins (matches lax.top_k ordering).
// ---------------------------------------------------------------------------
__global__ __launch_bounds__(256) void topk_kernel(
    const float* __restrict__ logits,
    int* __restrict__ top_idx, float* __restrict__ top_s)
{
    __shared__ float cv[256 * TOPK];   // 16 KB candidate values
    __shared__ int   ci[256 * TOPK];   // 16 KB candidate indices
    __shared__ float rm[256];
    __shared__ float rs[256];

    const int h   = blockIdx.x;
    const int tid = threadIdx.x;
    const float scale = 0.17677669529663687f;   // 1/sqrt(32)
    const float* L = logits + (size_t)h * NROWS;

    float lv[TOPK]; int li[TOPK];
    #pragma unroll
    for (int i = 0; i < TOPK; ++i) { lv[i] = -INFINITY; li[i] = 0x7fffffff; }

    float m = -INFINITY, s = 0.0f;
    for (int i = tid; i < NROWS; i += 256) {
        const float x = L[i] * scale;
        if (x > m) { s = s * __expf(m - x) + 1.0f; m = x; }
        else       { s += __expf(x - m); }
        if (x > lv[TOPK - 1]) {
            int p = TOPK - 1;
            while (p > 0 && (x > lv[p - 1])) {
                lv[p] = lv[p - 1]; li[p] = li[p - 1]; --p;
            }
            // equal values: keep earlier (lower-index) entry ahead
            while (p > 0 && x == lv[p - 1] && i < li[p - 1]) {
                lv[p] = lv[p - 1]; li[p] = li[p - 1]; --p;
            }
            lv[p] = x; li[p] = i;
        }
    }

    #pragma unroll
    for (int i = 0; i < TOPK; ++i) { cv[tid * TOPK + i] = lv[i]; ci[tid * TOPK + i] = li[i]; }
    rm[tid] = m; rs[tid] = s;
    __syncthreads();

    for (int stride = 128; stride > 0; stride >>= 1) {
        if (tid < stride) {
            // merge softmax stats
            const float m1 = rm[tid], m2 = rm[tid + stride];
            const float M  = fmaxf(m1, m2);
            rs[tid] = rs[tid] * __expf(m1 - M) + rs[tid + stride] * __expf(m2 - M);
            rm[tid] = M;
            // merge two sorted-descending 16-lists -> top 16
            float* a  = cv + tid * TOPK;            int* ai = ci + tid * TOPK;
            float* b  = cv + (tid + stride) * TOPK; int* bi = ci + (tid + stride) * TOPK;
            float ov[TOPK]; int oi[TOPK];
            int pa = 0, pb = 0;
            #pragma unroll
            for (int o = 0; o < TOPK; ++o) {
                const float av = a[pa], bv = b[pb];
                const bool takeA = (av > bv) || (av == bv && ai[pa] <= bi[pb]);
                if (takeA) { ov[o] = av; oi[o] = ai[pa]; ++pa; }
                else       { ov[o] = bv; oi[o] = bi[pb]; ++pb; }
            }
            #pragma unroll
            for (int o = 0; o < TOPK; ++o) { a[o] = ov[o]; ai[o] = oi[o]; }
        }
        __syncthreads();
    }

    if (tid < TOPK) {
        const float M = rm[0], S = rs[0];
        top_idx[h * TOPK + tid] = ci[tid];
        top_s  [h * TOPK + tid] = __expf(cv[tid] - M) / S;
    }
}

// ---------------------------------------------------------------------------
// Kernel 3: lazy V at the 128 selected rows, weighted sum, output projection.
// thread j = h*32 + d': concat[j] = sum_k s_k*(Wv[j]·e[idx_k] + bv[j])
// then out[o] = Wo[o]·concat + bo[o].
// ---------------------------------------------------------------------------
__global__ __launch_bounds__(256) void output_kernel(
    const float* __restrict__ E,  const float* __restrict__ Wv,
    const float* __restrict__ bv, const float* __restrict__ Wo,
    const float* __restrict__ bo, const int* __restrict__ top_idx,
    const float* __restrict__ top_s, float* __restrict__ out)
{
    __shared__ float concat[256];
    __shared__ int   sidx[NHEADS * TOPK];
    __shared__ float ssc [NHEADS * TOPK];

    const int tid = threadIdx.x;
    if (tid < NHEADS * TOPK) { sidx[tid] = top_idx[tid]; ssc[tid] = top_s[tid]; }
    __syncthreads();

    const int h = tid >> 5;
    const float* wvr = Wv + (size_t)tid * DCOLS;
    float acc = 0.0f, ssum = 0.0f;
    for (int k = 0; k < TOPK; ++k) {
        const int   idx = sidx[h * TOPK + k];
        const float sc  = ssc [h * TOPK + k];
        const float* er = E + (size_t)idx * DCOLS;
        float d = 0.0f;
        #pragma unroll 4
        for (int c = 0; c < DCOLS; ++c) d += wvr[c] * er[c];
        acc  += sc * d;
        ssum += sc;
    }
    concat[tid] = acc + ssum * bv[tid];
    __syncthreads();

    const float* wor = Wo + (size_t)tid * DCOLS;
    float o = bo[tid];
    #pragma unroll 4
    for (int c = 0; c < DCOLS; ++c) o += wor[c] * concat[c];
    out[tid] = o;
}

// ---------------------------------------------------------------------------
// launch
// ---------------------------------------------------------------------------
extern "C" void kernel_launch(void* const* d_in, const int* in_sizes, int n_in,
                              void* d_out, int out_size, void* d_ws, size_t ws_size,
                              hipStream_t stream) {
    const float* ce = (const float*)d_in[0];   // class_embedding (256)
    const float* E  = (const float*)d_in[1];   // entity_embeddings (N,256)
    const float* Wq = (const float*)d_in[2];
    const float* bq = (const float*)d_in[3];
    const float* Wk = (const float*)d_in[4];
    // d_in[5] = bk: constant per head inside softmax -> cancels, unused
    const float* Wv = (const float*)d_in[6];
    const float* bv = (const float*)d_in[7];
    const float* Wo = (const float*)d_in[8];
    const float* bo = (const float*)d_in[9];
    float* out = (float*)d_out;

    // workspace layout
    char* ws = (char*)d_ws;
    float* wfoldB  = (float*)(ws);                 // 256*16 floats = 16 KB
    int*   top_idx = (int*)  (ws + 16384);         // 8*16 ints
    float* top_s   = (float*)(ws + 16384 + 512);   // 8*16 floats
    float* logits  = (float*)(ws + 32768);         // 8*200000 floats = 6.4 MB

    fold_kernel<<<1, 256, 0, stream>>>(ce, Wq, bq, Wk, wfoldB);

    const int blocksA = (NTILES + 7) / 8;          // 8 waves/block, 1 tile/wave
    logits_wmma_kernel<<<blocksA, 256, 0, stream>>>(E, wfoldB, logits);

    topk_kernel<<<NHEADS, 256, 0, stream>>>(logits, top_idx, top_s);

    output_kernel<<<1, 256, 0, stream>>>(E, Wv, bv, Wo, bo, top_idx, top_s, out);
}